// LightGCN_87557203296880
// MI455X (gfx1250) — compile-verified
//
#include <hip/hip_runtime.h>

#define THREADS 256
#define EDGES_PER_TILE 256
#define TILES_PER_BLOCK 8
#define SCAN_CHUNK 1024          // elements per scan block (256 threads x 4)
#define NODES_PER_BLOCK (THREADS / 32)

// ---------------------------------------------------------------------------
// CDNA5 async global->LDS staging (gfx1250): ASYNCcnt-tracked copy.
// ---------------------------------------------------------------------------
__device__ __forceinline__ void async_load_b32_to_lds(void* lds_ptr, const void* gptr) {
  unsigned lds_addr = (unsigned)(unsigned long long)lds_ptr;  // LDS offset = addr[31:0]
  unsigned long long ga = (unsigned long long)gptr;
  asm volatile("global_load_async_to_lds_b32 %0, %1, off"
               :: "v"(lds_addr), "v"(ga)
               : "memory");
}

template <int N>
__device__ __forceinline__ void wait_asynccnt() {
#if __has_builtin(__builtin_amdgcn_s_wait_asynccnt)
  __builtin_amdgcn_s_wait_asynccnt(N);
#else
  asm volatile("s_wait_asynccnt %0" :: "i"(N) : "memory");
#endif
}

// ---------------------------------------------------------------------------
__global__ __launch_bounds__(THREADS) void zero_i32(int* __restrict__ p, int n) {
  int i = blockIdx.x * THREADS + threadIdx.x;
  if (i < n) p[i] = 0;
}

// cnt[d] = in-degree of node d
__global__ __launch_bounds__(THREADS) void edge_count(
    const int* __restrict__ dst, int* __restrict__ cnt, int n_edges) {
  int e = blockIdx.x * THREADS + threadIdx.x;
  if (e < n_edges) atomicAdd(&cnt[dst[e]], 1);
}

// Per-block sums of cnt (SCAN_CHUNK elements per block).
__global__ __launch_bounds__(THREADS) void scan_reduce(
    const int* __restrict__ cnt, int* __restrict__ bsum, int n) {
  __shared__ int sdata[THREADS];
  const int t = threadIdx.x;
  const int base = blockIdx.x * SCAN_CHUNK + t * 4;
  int s = 0;
#pragma unroll
  for (int k = 0; k < 4; ++k) {
    int j = base + k;
    if (j < n) s += cnt[j];
  }
  sdata[t] = s;
  __syncthreads();
  for (int off = THREADS / 2; off > 0; off >>= 1) {
    if (t < off) sdata[t] += sdata[t + off];
    __syncthreads();
  }
  if (t == 0) bsum[blockIdx.x] = sdata[0];
}

// Single-block exclusive scan of block sums (nb <= 1024); writes total edges
// to rowptr_last.
__global__ __launch_bounds__(THREADS) void scan_blocksums(
    const int* __restrict__ bsum, int* __restrict__ boff,
    int* __restrict__ rowptr_last, int nb) {
  __shared__ int lds[1024];
  for (int j = threadIdx.x; j < nb; j += THREADS) lds[j] = bsum[j];
  __syncthreads();
  if (threadIdx.x == 0) {
    int run = 0;
    for (int j = 0; j < nb; ++j) { int c = lds[j]; lds[j] = run; run += c; }
    *rowptr_last = run;
  }
  __syncthreads();
  for (int j = threadIdx.x; j < nb; j += THREADS) boff[j] = lds[j];
}

// Per-block exclusive scan with global block offset; writes rowptr + cursor.
__global__ __launch_bounds__(THREADS) void scan_offsets(
    const int* __restrict__ cnt, const int* __restrict__ boff,
    int* __restrict__ rowptr, int* __restrict__ cursor, int n) {
  __shared__ int sthread[THREADS];
  __shared__ int sscan[THREADS];
  const int t = threadIdx.x;
  const int base = blockIdx.x * SCAN_CHUNK + t * 4;
  int c[4];
  int ts = 0;
#pragma unroll
  for (int k = 0; k < 4; ++k) {
    c[k] = (base + k < n) ? cnt[base + k] : 0;
    ts += c[k];
  }
  sthread[t] = ts;
  __syncthreads();
  if (t == 0) {
    int run = 0;
    for (int j = 0; j < THREADS; ++j) { sscan[j] = run; run += sthread[j]; }
  }
  __syncthreads();
  int run = boff[blockIdx.x] + sscan[t];
#pragma unroll
  for (int k = 0; k < 4; ++k) {
    int j = base + k;
    if (j < n) { rowptr[j] = run; cursor[j] = run; run += c[k]; }
  }
}

// ---------------------------------------------------------------------------
// Scatter edges into dst-sorted order. Edge tiles double-buffered through LDS
// via CDNA5 async global->LDS copies (overlaps HBM edge stream with cursor
// atomics). Staging is block-uniform so s_wait_asynccnt 3 retires tile t.
// ---------------------------------------------------------------------------
__global__ __launch_bounds__(THREADS) void edge_scatter(
    const int* __restrict__ src, const int* __restrict__ dst,
    const float* __restrict__ val, int* __restrict__ cursor,
    int* __restrict__ ssrc, float* __restrict__ sval, int n_edges) {
  __shared__ int   s_src[2][EDGES_PER_TILE];
  __shared__ int   s_dst[2][EDGES_PER_TILE];
  __shared__ float s_val[2][EDGES_PER_TILE];

  const int tid = threadIdx.x;
  const long long last = (long long)n_edges - 1;
  const long long block_base =
      (long long)blockIdx.x * (EDGES_PER_TILE * TILES_PER_BLOCK);

  {
    long long e = block_base + tid;
    if (e > last) e = last;
    async_load_b32_to_lds(&s_src[0][tid], src + e);
    async_load_b32_to_lds(&s_dst[0][tid], dst + e);
    async_load_b32_to_lds(&s_val[0][tid], val + e);
  }

  for (int t = 0; t < TILES_PER_BLOCK; ++t) {
    const long long tile_base = block_base + (long long)t * EDGES_PER_TILE;
    if (tile_base > last) break;  // block-uniform
    const int cur = t & 1, nxt = cur ^ 1;

    const long long next_base = tile_base + EDGES_PER_TILE;
    if ((t + 1 < TILES_PER_BLOCK) && (next_base <= last)) {
      long long e = next_base + tid;
      if (e > last) e = last;
      async_load_b32_to_lds(&s_src[nxt][tid], src + e);
      async_load_b32_to_lds(&s_dst[nxt][tid], dst + e);
      async_load_b32_to_lds(&s_val[nxt][tid], val + e);
      wait_asynccnt<3>();
    } else {
      wait_asynccnt<0>();
    }
    __syncthreads();

    const long long ge = tile_base + tid;
    if (ge <= last) {
      const int d = s_dst[cur][tid];
      const int p = atomicAdd(&cursor[d], 1);
      ssrc[p] = s_src[cur][tid];
      sval[p] = s_val[cur][tid];
    }
    __syncthreads();
  }
}

// ---------------------------------------------------------------------------
// h0 = acc = concat(user_emb, item_emb)
// ---------------------------------------------------------------------------
__global__ __launch_bounds__(THREADS) void lgcn_init(
    const float* __restrict__ ue, const float* __restrict__ ie,
    float* __restrict__ h0, float* __restrict__ acc,
    long long n_user_elems, long long total_elems) {
  long long i = ((long long)blockIdx.x * THREADS + threadIdx.x) * 4;
  if (i >= total_elems) return;
  float4 v;
  if (i < n_user_elems) {
    v = *reinterpret_cast<const float4*>(ue + i);
  } else {
    v = *reinterpret_cast<const float4*>(ie + (i - n_user_elems));
  }
  *reinterpret_cast<float4*>(h0 + i) = v;
  *reinterpret_cast<float4*>(acc + i) = v;
}

// ---------------------------------------------------------------------------
// CSR segment-reduce SpMM with fused accumulator epilogue, zero atomics.
// One wave32 per dst node, 2 dims per lane. Chunks of 32 (src,val) pairs are
// register double-buffered: chunk k+1's coalesced loads issue before chunk k
// is consumed, overlapping the edge stream with readlane-broadcast gathers
// (saddr-form, 256B rows, L2-resident table). Epilogue:
//   acc[d] = (acc[d] + out) * scale ;  y[d] = out (skipped on last layer).
// ---------------------------------------------------------------------------
__global__ __launch_bounds__(THREADS) void lgcn_spmm_csr(
    const int* __restrict__ rowptr, const int* __restrict__ ssrc,
    const float* __restrict__ sval, const float* __restrict__ x,
    float* __restrict__ y, float* __restrict__ acc,
    float scale, int store_h, int n_nodes) {
  const int wave = threadIdx.x >> 5;
  const int lane = threadIdx.x & 31;
  const int d = blockIdx.x * NODES_PER_BLOCK + wave;
  if (d >= n_nodes) return;

  const int start = rowptr[d];
  const int end   = rowptr[d + 1];
  const int col   = lane << 1;

  float ax = 0.f, ay = 0.f;

  int   es = 0;
  float ev = 0.f;
  if (start < end) {
    int idx = start + lane;
    if (idx >= end) idx = end - 1;
    es = ssrc[idx];
    ev = sval[idx];
  }
  for (int base = start; base < end; base += 32) {
    const int n = min(32, end - base);
    const int   cur_es = es;
    const float cur_ev = ev;
    const int next = base + 32;
    if (next < end) {  // issue next chunk's loads before consuming current
      int idx = next + lane;
      if (idx >= end) idx = end - 1;
      es = ssrc[idx];
      ev = sval[idx];
    }
    for (int i = 0; i < n; ++i) {
      const int   s = __builtin_amdgcn_readlane(cur_es, i);
      const float v = __int_as_float(__builtin_amdgcn_readlane(__float_as_int(cur_ev), i));
      const float2 xv = *reinterpret_cast<const float2*>(x + ((long long)s << 6) + col);
      ax += v * xv.x;
      ay += v * xv.y;
    }
  }

  const long long row = (long long)d << 6;
  float* arow = acc + row + col;
  float2 a = *reinterpret_cast<float2*>(arow);
  a.x = (a.x + ax) * scale;
  a.y = (a.y + ay) * scale;
  *reinterpret_cast<float2*>(arow) = a;
  if (store_h) {
    float2 o;
    o.x = ax;
    o.y = ay;
    *reinterpret_cast<float2*>(y + row + col) = o;
  }
}

// ---------------------------------------------------------------------------
extern "C" void kernel_launch(void* const* d_in, const int* in_sizes, int n_in,
                              void* d_out, int out_size, void* d_ws, size_t ws_size,
                              hipStream_t stream) {
  const float* ue   = (const float*)d_in[0];
  const float* ie   = (const float*)d_in[1];
  const int*   esrc = (const int*)d_in[2];
  const int*   edst = (const int*)d_in[3];
  const float* eval = (const float*)d_in[4];
  float* acc = (float*)d_out;

  const long long n_user_elems = in_sizes[0];
  const long long total = (long long)in_sizes[0] + (long long)in_sizes[1];
  const int n_edges = in_sizes[2];
  const int n_nodes = (int)(total >> 6);  // d = 64

  // ---- workspace layout (4-element aligned regions) ----
  auto align4 = [](long long v) { return (v + 3) & ~3LL; };
  float* hA     = (float*)d_ws;                 // total floats
  float* hB     = hA + total;                   // total floats
  int*   ssrc   = (int*)(hB + total);           // n_edges
  float* sval   = (float*)(ssrc + align4(n_edges));  // n_edges
  int*   rowptr = (int*)(sval + align4(n_edges));    // n_nodes + 1
  int*   cursor = rowptr + align4(n_nodes + 1);      // n_nodes
  int*   cnt    = cursor + align4(n_nodes);          // n_nodes
  int*   bsum   = cnt + align4(n_nodes);             // <= 1024
  int*   boff   = bsum + 1024;                       // <= 1024

  const int nb = (n_nodes + SCAN_CHUNK - 1) / SCAN_CHUNK;
  const int vec_blocks = (int)((total / 4 + THREADS - 1) / THREADS);
  const long long epb = (long long)EDGES_PER_TILE * TILES_PER_BLOCK;
  const int edge_blocks      = (int)(((long long)n_edges + epb - 1) / epb);
  const int edge_blocks_flat = (n_edges + THREADS - 1) / THREADS;
  const int node_blocks      = (n_nodes + THREADS - 1) / THREADS;
  const int spmm_blocks      = (n_nodes + NODES_PER_BLOCK - 1) / NODES_PER_BLOCK;

  // ---- build dst-sorted CSR once (reused by all 3 layers) ----
  zero_i32<<<node_blocks, THREADS, 0, stream>>>(cnt, n_nodes);
  edge_count<<<edge_blocks_flat, THREADS, 0, stream>>>(edst, cnt, n_edges);
  scan_reduce<<<nb, THREADS, 0, stream>>>(cnt, bsum, n_nodes);
  scan_blocksums<<<1, THREADS, 0, stream>>>(bsum, boff, rowptr + n_nodes, nb);
  scan_offsets<<<nb, THREADS, 0, stream>>>(cnt, boff, rowptr, cursor, n_nodes);
  edge_scatter<<<edge_blocks, THREADS, 0, stream>>>(esrc, edst, eval, cursor,
                                                    ssrc, sval, n_edges);

  // ---- LightGCN propagation (acc update fused into SpMM epilogue) ----
  lgcn_init<<<vec_blocks, THREADS, 0, stream>>>(ue, ie, hA, acc, n_user_elems, total);
  for (int L = 0; L < 3; ++L) {
    const float scale = (L == 2) ? 0.25f : 1.0f;
    const int store_h = (L < 2) ? 1 : 0;  // last layer's h is never read again
    lgcn_spmm_csr<<<spmm_blocks, THREADS, 0, stream>>>(rowptr, ssrc, sval,
                                                       hA, hB, acc,
                                                       scale, store_h, n_nodes);
    float* t = hA; hA = hB; hB = t;
  }
}